// Attention_87694642250332
// MI455X (gfx1250) — compile-verified
//
#include <hip/hip_runtime.h>
#include <hip/hip_bf16.h>

// Attention: out = softmax(clip(QK^T/8) masked * decay) @ V, also emits p_attn.
// B=2 H=8 S=2048 D=64, all f32.  Memory-bound (~0.55GB traffic -> ~24us @23.3TB/s),
// so: f32 WMMA (16x16x4), full 16x2048 score block resident in CDNA5's 320KB LDS
// (single pass over the 256MB decay/p_attn tensors), NT hints on the streams.

typedef __attribute__((ext_vector_type(2))) float v2f;
typedef __attribute__((ext_vector_type(8))) float v8f;

#define B_ 2
#define H_ 8
#define S_ 2048
#define D_ 64
#define QB 16                 // query rows per workgroup
#define NWAVE 4
#define NTHREADS (NWAVE * 32)
#define KTILES (S_ / 16)      // 128 key tiles of 16
#define LST 2050              // LDS score row stride (floats): 8B-aligned, bank-rotated

__global__ __launch_bounds__(NTHREADS)
void Attention_87694642250332_kernel(const float* __restrict__ Q,
                                     const float* __restrict__ K,
                                     const float* __restrict__ V,
                                     const unsigned char* __restrict__ M,  // jnp.bool_ -> 1 byte
                                     const float* __restrict__ W,          // decay
                                     float* __restrict__ out,
                                     float* __restrict__ p) {
  extern __shared__ float smem[];
  float* sc     = smem;                 // [QB][LST] raw masked/decayed scores
  float* outAcc = sc + QB * LST;        // [QB][D_]
  float* red    = outAcc + QB * D_;     // [QB][8] reduction scratch
  float* rowMax = red + QB * 8;         // [QB]
  float* rowInv = rowMax + QB;          // [QB]

  const int tid    = threadIdx.x;
  const int wave   = tid >> 5;
  const int lane   = tid & 31;
  const int laneLo = lane & 15;
  const int hiSel  = lane >> 4;         // 0: K={4c,4c+1}, 1: K={4c+2,4c+3}

  const int bid  = blockIdx.x;
  const int qblk = bid % (S_ / QB);
  const int h    = (bid / (S_ / QB)) % H_;
  const int b    = bid / ((S_ / QB) * H_);
  const size_t bh = (size_t)b * H_ + h;
  const int q0   = qblk * QB;

  // ---- Q tile in WMMA f32 16x4 A-layout registers: aQ[c] covers K dims 4c..4c+3 ----
  v2f aQ[16];
  {
    const float* qrow = Q + (bh * S_ + (size_t)(q0 + laneLo)) * D_ + 2 * hiSel;
#pragma unroll
    for (int c = 0; c < 16; ++c) aQ[c] = *(const v2f*)(qrow + 4 * c);
  }

  // ---- Phase 1: scores = clip(QK^T*0.125) masked * decay -> LDS ----
  for (int kt = wave; kt < KTILES; kt += NWAVE) {
    const float* krow = K + (bh * S_ + (size_t)(kt * 16 + laneLo)) * D_ + 2 * hiSel;
    v8f acc = {};
#pragma unroll
    for (int cc = 0; cc < 16; ++cc) {
      v2f bK = *(const v2f*)(krow + 4 * cc);
      acc = __builtin_amdgcn_wmma_f32_16x16x4_f32(false, aQ[cc], false, bK,
                                                  (short)0, acc, false, false);
    }
    const int kg = kt * 16 + laneLo;
    if (kt + NWAVE < KTILES)  // stream-ahead on the 256MB decay tensor
      __builtin_prefetch(W + (bh * S_ + (size_t)(q0 + laneLo)) * S_ +
                             (size_t)(kt + NWAVE) * 16 + laneLo, 0, 1);
#pragma unroll
    for (int v = 0; v < 8; ++v) {
      const int mm = v + 8 * hiSel;         // C layout: VGPR v -> M=v (+8 for hi lanes)
      const int qg = q0 + mm;
      float s = acc[v] * 0.125f;            // 1/sqrt(64)
      s = fminf(fmaxf(s, 1e-9f), 1e9f);
      const unsigned char mk = M[((size_t)b * S_ + qg) * S_ + kg];
      s = mk ? s : -1e9f;
      s *= __builtin_nontemporal_load(W + (bh * S_ + (size_t)qg) * S_ + kg);
      sc[mm * LST + kg] = s;
    }
  }

  for (int i = tid; i < QB * D_; i += NTHREADS) outAcc[i] = 0.0f;
  __syncthreads();

  // ---- Phase 2a: exact softmax stats (row max, row sum) ----
  const int tRow = tid >> 3;                // 8 threads per row
  const int tSub = tid & 7;
  {
    float lmax = -3.4e38f;
    for (int k = tSub; k < S_; k += 8) lmax = fmaxf(lmax, sc[tRow * LST + k]);
    red[tRow * 8 + tSub] = lmax;
  }
  __syncthreads();
  if (tid < QB) {
    float mx = red[tid * 8];
#pragma unroll
    for (int i = 1; i < 8; ++i) mx = fmaxf(mx, red[tid * 8 + i]);
    rowMax[tid] = mx;
  }
  __syncthreads();
  {
    const float mrow = rowMax[tRow];
    float lsum = 0.0f;
    for (int k = tSub; k < S_; k += 8) lsum += __expf(sc[tRow * LST + k] - mrow);
    red[tRow * 8 + tSub] = lsum;
  }
  __syncthreads();
  if (tid < QB) {
    float sm = 0.0f;
#pragma unroll
    for (int i = 0; i < 8; ++i) sm += red[tid * 8 + i];
    rowInv[tid] = 1.0f / sm;
  }
  __syncthreads();

  // ---- Phase 2b: stream p_attn out (coalesced 512B/instr, non-temporal) ----
  for (int idx = tid; idx < QB * S_; idx += NTHREADS) {
    const int mm = idx >> 11;               // S_ == 2048
    const int kk = idx & (S_ - 1);
    const float pp = __expf(sc[mm * LST + kk] - rowMax[mm]) * rowInv[mm];
    __builtin_nontemporal_store(pp, p + (bh * S_ + (size_t)(q0 + mm)) * S_ + kk);
  }

  // ---- Phase 2c: out = P @ V via WMMA; waves split the K range, LDS-atomic reduce ----
  {
    v8f acc[4] = {};                        // 4 N-tiles cover D=64
    const float mrow = rowMax[laneLo];      // A rows: M = laneLo for both lane halves
    const float irow = rowInv[laneLo];
    for (int kt = wave * 32; kt < wave * 32 + 32; ++kt) {
      v2f aP[4];
#pragma unroll
      for (int c = 0; c < 4; ++c) {         // A-layout p, recomputed from LDS scores
        const int kk = kt * 16 + 4 * c + 2 * hiSel;
        v2f t;
        t.x = __expf(sc[laneLo * LST + kk    ] - mrow) * irow;
        t.y = __expf(sc[laneLo * LST + kk + 1] - mrow) * irow;
        aP[c] = t;
      }
#pragma unroll
      for (int nt = 0; nt < 4; ++nt) {
#pragma unroll
        for (int c = 0; c < 4; ++c) {       // B[k][n] = V[key][d]
          const float* vb = V + (bh * S_ + (size_t)(kt * 16 + 4 * c + 2 * hiSel)) * D_ +
                            nt * 16 + laneLo;
          v2f bV; bV.x = vb[0]; bV.y = vb[D_];
          acc[nt] = __builtin_amdgcn_wmma_f32_16x16x4_f32(false, aP[c], false, bV,
                                                          (short)0, acc[nt], false, false);
        }
      }
    }
#pragma unroll
    for (int nt = 0; nt < 4; ++nt)
#pragma unroll
      for (int v = 0; v < 8; ++v)
        atomicAdd(&outAcc[(v + 8 * hiSel) * D_ + nt * 16 + laneLo], acc[nt][v]);  // ds_add_f32
  }
  __syncthreads();

  // ---- write out tile (coalesced) ----
  for (int idx = tid; idx < QB * D_; idx += NTHREADS) {
    const int mm = idx >> 6;
    const int dd = idx & (D_ - 1);
    out[(bh * S_ + (size_t)(q0 + mm)) * D_ + dd] = outAcc[idx];
  }
}

extern "C" void kernel_launch(void* const* d_in, const int* in_sizes, int n_in,
                              void* d_out, int out_size, void* d_ws, size_t ws_size,
                              hipStream_t stream) {
  (void)in_sizes; (void)n_in; (void)d_ws; (void)ws_size; (void)out_size;
  const float* Q = (const float*)d_in[0];
  const float* K = (const float*)d_in[1];
  const float* V = (const float*)d_in[2];
  const unsigned char* M = (const unsigned char*)d_in[3];
  const float* W = (const float*)d_in[4];
  float* out = (float*)d_out;
  float* p   = out + (size_t)B_ * H_ * S_ * D_;   // tuple: (out, p_attn) flat-concatenated

  const int nblocks = B_ * H_ * (S_ / QB);        // 2048 workgroups
  const size_t smem = (size_t)(QB * LST + QB * D_ + QB * 8 + QB + QB) * sizeof(float); // ~136KB
  Attention_87694642250332_kernel<<<nblocks, NTHREADS, smem, stream>>>(Q, K, V, M, W, out, p);
}